// Net_1846835937364
// MI455X (gfx1250) — compile-verified
//
#include <hip/hip_runtime.h>
#include <hip/hip_bf16.h>
#include <math.h>

// ---- problem constants (from reference) ----
#define NFEAT  128
#define NHID   64
#define NCLASS 40

typedef __attribute__((ext_vector_type(2))) float v2f;
typedef __attribute__((ext_vector_type(8))) float v8f;

// ---------------------------------------------------------------------------
// Degree / normalization
// ---------------------------------------------------------------------------
__global__ void init_deg_kernel(float* __restrict__ deg, int n) {
    int i = blockIdx.x * blockDim.x + threadIdx.x;
    if (i < n) deg[i] = 1.0f;  // self-loop contributes 1
}

__global__ void degree_kernel(const long long* __restrict__ src,
                              float* __restrict__ deg, long long nedges) {
    long long i = (long long)blockIdx.x * blockDim.x + threadIdx.x;
    if (i < nedges) atomicAdd(&deg[(int)src[i]], 1.0f);
}

__global__ void dinv_kernel(float* __restrict__ deg, int n) {
    int i = blockIdx.x * blockDim.x + threadIdx.x;
    if (i < n) deg[i] = rsqrtf(deg[i]);  // in place: deg -> deg^-0.5
}

// ---------------------------------------------------------------------------
// GEMM1: H1[n,64] = X[n,128] @ W1[128,64] + b1   (fp32 WMMA 16x16x4)
// block = 256 threads = 8 waves; each wave owns one 16-row tile and computes
// ALL four 16-col tiles: A loaded once per K-step, 4 back-to-back WMMAs.
// ---------------------------------------------------------------------------
__global__ void gemm1_wmma(const float* __restrict__ X, const float* __restrict__ W,
                           const float* __restrict__ bias, float* __restrict__ H,
                           int nnodes) {
    const int wave = threadIdx.x >> 5;
    const int lane = threadIdx.x & 31;
    const int half = lane >> 4;       // 0: K=0,1   1: K=2,3
    const int l16  = lane & 15;
    const int m0   = (blockIdx.x * 8 + wave) * 16;
    if (m0 >= nnodes) return;                            // uniform per wave

    const int arow = min(m0 + l16, nnodes - 1);          // A: lane = row M
    const float* __restrict__ xrow = X + (size_t)arow * NFEAT;

    v8f c0 = {}, c1 = {}, c2 = {}, c3 = {};
    for (int k = 0; k < NFEAT; k += 4) {
        const int ka = k + half * 2;
        const v2f a = *(const v2f*)(xrow + ka);          // A[M, ka..ka+1]
        const float* __restrict__ w0 = W + (size_t)ka * NHID + l16;
        const float* __restrict__ w1 = w0 + NHID;
        v2f b;
        b.x = w0[0];  b.y = w1[0];
        c0 = __builtin_amdgcn_wmma_f32_16x16x4_f32(false, a, false, b, (short)0, c0, false, false);
        b.x = w0[16]; b.y = w1[16];
        c1 = __builtin_amdgcn_wmma_f32_16x16x4_f32(false, a, false, b, (short)0, c1, false, false);
        b.x = w0[32]; b.y = w1[32];
        c2 = __builtin_amdgcn_wmma_f32_16x16x4_f32(false, a, false, b, (short)0, c2, false, false);
        b.x = w0[48]; b.y = w1[48];
        c3 = __builtin_amdgcn_wmma_f32_16x16x4_f32(false, a, false, b, (short)0, c3, false, false);
    }

    const float bv0 = bias[l16];
    const float bv1 = bias[l16 + 16];
    const float bv2 = bias[l16 + 32];
    const float bv3 = bias[l16 + 48];

    if (m0 + 16 <= nnodes) {                             // uniform: full tile
        #pragma unroll
        for (int r = 0; r < 8; ++r) {                    // VGPR r: M = r + 8*half
            float* __restrict__ hrow = H + (size_t)(m0 + half * 8 + r) * NHID + l16;
            hrow[0]  = c0[r] + bv0;
            hrow[16] = c1[r] + bv1;
            hrow[32] = c2[r] + bv2;
            hrow[48] = c3[r] + bv3;
        }
    } else {                                             // tail tile (unused for 100000)
        #pragma unroll
        for (int r = 0; r < 8; ++r) {
            const int row = m0 + half * 8 + r;
            if (row < nnodes) {
                float* __restrict__ hrow = H + (size_t)row * NHID + l16;
                hrow[0]  = c0[r] + bv0;
                hrow[16] = c1[r] + bv1;
                hrow[32] = c2[r] + bv2;
                hrow[48] = c3[r] + bv3;
            }
        }
    }
}

// ---------------------------------------------------------------------------
// GEMM2: H2[n,40] = relu(A1[n,64]) @ W2[64,40] + b2  (ReLU fused into A loads)
// block = 256 = 8 waves; each wave: one 16-row tile x 48 padded cols (3 tiles)
// ---------------------------------------------------------------------------
__global__ void gemm2_wmma(const float* __restrict__ A1, const float* __restrict__ W,
                           const float* __restrict__ bias, float* __restrict__ H2,
                           int nnodes) {
    const int wave = threadIdx.x >> 5;
    const int lane = threadIdx.x & 31;
    const int half = lane >> 4;
    const int l16  = lane & 15;
    const int m0   = (blockIdx.x * 8 + wave) * 16;
    if (m0 >= nnodes) return;                            // uniform per wave

    const int arow = min(m0 + l16, nnodes - 1);
    const float* __restrict__ a1row = A1 + (size_t)arow * NHID;
    const bool c2ok = (l16 + 32 < NCLASS);               // cols 32..47: only <40

    v8f c0 = {}, c1 = {}, c2 = {};
    for (int k = 0; k < NHID; k += 4) {
        const int ka = k + half * 2;
        v2f a;
        a.x = fmaxf(a1row[ka],     0.0f);                // fused ReLU
        a.y = fmaxf(a1row[ka + 1], 0.0f);
        const float* __restrict__ w0 = W + (size_t)ka * NCLASS + l16;
        const float* __restrict__ w1 = w0 + NCLASS;
        v2f b;
        b.x = w0[0];  b.y = w1[0];
        c0 = __builtin_amdgcn_wmma_f32_16x16x4_f32(false, a, false, b, (short)0, c0, false, false);
        b.x = w0[16]; b.y = w1[16];
        c1 = __builtin_amdgcn_wmma_f32_16x16x4_f32(false, a, false, b, (short)0, c1, false, false);
        b.x = c2ok ? w0[32] : 0.0f;
        b.y = c2ok ? w1[32] : 0.0f;
        c2 = __builtin_amdgcn_wmma_f32_16x16x4_f32(false, a, false, b, (short)0, c2, false, false);
    }

    const float bv0 = bias[l16];
    const float bv1 = bias[l16 + 16];
    const float bv2 = c2ok ? bias[l16 + 32] : 0.0f;

    if (m0 + 16 <= nnodes) {                             // uniform: full tile
        #pragma unroll
        for (int r = 0; r < 8; ++r) {
            float* __restrict__ hrow = H2 + (size_t)(m0 + half * 8 + r) * NCLASS + l16;
            hrow[0]  = c0[r] + bv0;
            hrow[16] = c1[r] + bv1;
            if (c2ok) hrow[32] = c2[r] + bv2;
        }
    } else {
        #pragma unroll
        for (int r = 0; r < 8; ++r) {
            const int row = m0 + half * 8 + r;
            if (row < nnodes) {
                float* __restrict__ hrow = H2 + (size_t)row * NCLASS + l16;
                hrow[0]  = c0[r] + bv0;
                hrow[16] = c1[r] + bv1;
                if (c2ok) hrow[32] = c2[r] + bv2;
            }
        }
    }
}

// ---------------------------------------------------------------------------
// Self-loop initialization: acc[n,f] = dinv[n]^2 * h[n,f]  (float4 vectorized)
// ---------------------------------------------------------------------------
template <int F>
__global__ void selfloop_init_kernel(const float* __restrict__ dinv,
                                     const float* __restrict__ h,
                                     float* __restrict__ acc, int nnodes) {
    constexpr int Q = F / 4;
    long long idx = (long long)blockIdx.x * blockDim.x + threadIdx.x;
    long long total = (long long)nnodes * Q;
    if (idx >= total) return;
    int node = (int)(idx / Q);
    float di = dinv[node];
    float sc = di * di;
    const float4 v = *(const float4*)(h + idx * 4);
    float4 o; o.x = sc * v.x; o.y = sc * v.y; o.z = sc * v.z; o.w = sc * v.w;
    *(float4*)(acc + idx * 4) = o;
}

// ---------------------------------------------------------------------------
// Edge aggregation: acc[dst,f] += dinv[src]*dinv[dst] * h[src,f]
// One thread per (edge, 4 features): float4 gather + 4 f32 atomics into L2.
// ---------------------------------------------------------------------------
template <int F>
__global__ void aggregate_kernel(const long long* __restrict__ src,
                                 const long long* __restrict__ dst,
                                 const float* __restrict__ dinv,
                                 const float* __restrict__ h,
                                 float* __restrict__ acc, long long nedges) {
    constexpr int Q = F / 4;
    long long idx = (long long)blockIdx.x * blockDim.x + threadIdx.x;
    long long total = nedges * Q;
    if (idx >= total) return;
    long long e = idx / Q;
    int q = (int)(idx - e * Q);
    int s = (int)src[e];
    int d = (int)dst[e];
    float norm = dinv[s] * dinv[d];
    const float4 v = *(const float4*)(h + (size_t)s * F + q * 4);
    float* __restrict__ ap = acc + (size_t)d * F + q * 4;
    atomicAdd(ap + 0, norm * v.x);
    atomicAdd(ap + 1, norm * v.y);
    atomicAdd(ap + 2, norm * v.z);
    atomicAdd(ap + 3, norm * v.w);
}

// ---------------------------------------------------------------------------
// Row-wise log_softmax over NCLASS=40, in place. One wave32 per node.
// ---------------------------------------------------------------------------
__global__ void logsoftmax_kernel(float* __restrict__ out, int nnodes) {
    int gwave = (int)(((long long)blockIdx.x * blockDim.x + threadIdx.x) >> 5);
    int lane = threadIdx.x & 31;
    if (gwave >= nnodes) return;
    float* row = out + (size_t)gwave * NCLASS;

    float v0 = row[lane];                                        // lane < 32 < 40
    float v1 = (lane + 32 < NCLASS) ? row[lane + 32] : -INFINITY;

    float m = fmaxf(v0, v1);
    #pragma unroll
    for (int off = 16; off > 0; off >>= 1)
        m = fmaxf(m, __shfl_xor(m, off, 32));

    float s = expf(v0 - m);
    if (lane + 32 < NCLASS) s += expf(v1 - m);
    #pragma unroll
    for (int off = 16; off > 0; off >>= 1)
        s += __shfl_xor(s, off, 32);

    float lse = m + logf(s);
    row[lane] = v0 - lse;
    if (lane + 32 < NCLASS) row[lane + 32] = v1 - lse;
}

// ---------------------------------------------------------------------------
// Host launcher
// ---------------------------------------------------------------------------
extern "C" void kernel_launch(void* const* d_in, const int* in_sizes, int n_in,
                              void* d_out, int out_size, void* d_ws, size_t ws_size,
                              hipStream_t stream) {
    const float*     x    = (const float*)d_in[0];
    const long long* eidx = (const long long*)d_in[1];   // int64 in reference
    const float*     W1   = (const float*)d_in[2];
    const float*     b1   = (const float*)d_in[3];
    const float*     W2   = (const float*)d_in[4];
    const float*     b2   = (const float*)d_in[5];
    float*           out  = (float*)d_out;

    const int       nnodes = in_sizes[0] / NFEAT;        // 100000
    const long long nedges = (long long)in_sizes[1] / 2; // 1600000
    const long long* src = eidx;
    const long long* dst = eidx + nedges;

    // workspace layout (floats): dinv | h1 | a1 | h2   (~67.6 MB, fits in L2)
    float* dinv = (float*)d_ws;
    float* h1   = dinv + nnodes;
    float* a1   = h1 + (size_t)nnodes * NHID;
    float* h2   = a1 + (size_t)nnodes * NHID;

    const int BLK = 256;
    const int mtiles16 = (nnodes + 15) / 16;             // 6250 row tiles
    const int gemmBlocks = (mtiles16 + 7) / 8;           // 8 waves/block

    // 1) degree -> dinv
    init_deg_kernel<<<(nnodes + BLK - 1) / BLK, BLK, 0, stream>>>(dinv, nnodes);
    degree_kernel<<<(unsigned)((nedges + BLK - 1) / BLK), BLK, 0, stream>>>(src, dinv, nedges);
    dinv_kernel<<<(nnodes + BLK - 1) / BLK, BLK, 0, stream>>>(dinv, nnodes);

    // 2) h1 = x @ W1 + b1  (fp32 WMMA)
    gemm1_wmma<<<gemmBlocks, 256, 0, stream>>>(x, W1, b1, h1, nnodes);

    // 3) a1 = dinv^2*h1 (self loops) + scatter of 1.6M edges
    {
        long long total = (long long)nnodes * (NHID / 4);
        selfloop_init_kernel<NHID><<<(unsigned)((total + BLK - 1) / BLK), BLK, 0, stream>>>(
            dinv, h1, a1, nnodes);
        long long etotal = nedges * (NHID / 4);
        aggregate_kernel<NHID><<<(unsigned)((etotal + BLK - 1) / BLK), BLK, 0, stream>>>(
            src, dst, dinv, h1, a1, nedges);
    }

    // 4) h2 = relu(a1) @ W2 + b2  (ReLU fused, fp32 WMMA)
    gemm2_wmma<<<gemmBlocks, 256, 0, stream>>>(a1, W2, b2, h2, nnodes);

    // 5) out = dinv^2*h2 + scatter of 1.6M edges  (accumulate directly in d_out)
    {
        long long total = (long long)nnodes * (NCLASS / 4);
        selfloop_init_kernel<NCLASS><<<(unsigned)((total + BLK - 1) / BLK), BLK, 0, stream>>>(
            dinv, h2, out, nnodes);
        long long etotal = nedges * (NCLASS / 4);
        aggregate_kernel<NCLASS><<<(unsigned)((etotal + BLK - 1) / BLK), BLK, 0, stream>>>(
            src, dst, dinv, h2, out, nedges);
    }

    // 6) log_softmax in place (one wave32 per node)
    {
        int waves_per_block = BLK / 32;
        int blocks = (nnodes + waves_per_block - 1) / waves_per_block;
        logsoftmax_kernel<<<blocks, BLK, 0, stream>>>(out, nnodes);
    }
}